// PeakAuxiliaryModule_2413771620531
// MI455X (gfx1250) — compile-verified
//
#include <hip/hip_runtime.h>
#include <hip/hip_bf16.h>
#include <stdint.h>

// ---------------- problem constants (match reference) ----------------
#define BATCH   8
#define LEN     4096
#define CFEAT   96          // total feature channels
#define DI      192         // d_inner
#define NST     16          // d_state
#define DTR     6           // dt_rank
#define BL      (BATCH*LEN) // 32768 rows
#define DBLP    48          // dbl padded cols (6 dt + 16 B + 16 C = 38 -> 48)
#define LNEPS   1e-5f

typedef __attribute__((ext_vector_type(2))) float v2f;
typedef __attribute__((ext_vector_type(8))) float v8f;

// =====================================================================
// WMMA f32 GEMM:  Y[M x N] = A[M x K] @ W[K x N]   (all row-major)
//
// Block = 8 waves = 128 rows x one 16-col N-tile. The shared W panel
// (K x 16 floats, K<=192) is staged once per block into LDS with
// GLOBAL_LOAD_ASYNC_TO_LDS_B128 (ASYNCcnt), then every wave reads its
// B-fragments from LDS (broadcast, conflict-free). K stepped by 4 via
// V_WMMA_F32_16X16X4_F32 (f32-exact, matches the reference precision).
//
// A-frag per ISA table: lanes 0-15 hold (K+0,K+1) of row M=lane,
//                       lanes 16-31 hold (K+2,K+3).
// B-frag assumed symmetric (K rows striped across VGPR x lane-half).
// ACC=1: Y += (residual accumulate, used for the W_out projection).
// =====================================================================
template<int ACC>
__global__ void wmma_gemm_f32_kernel(const float* __restrict__ A,
                                     const float* __restrict__ W,
                                     float* __restrict__ Y,
                                     int M, int N, int K, int nTilesN)
{
    __shared__ float shW[DI * 16];               // K-panel, row-major [K][16]

    const int tid  = threadIdx.x;
    const int wave = tid >> 5;
    const int lane = tid & 31;
    const int tN   = blockIdx.x % nTilesN;
    const int tM   = (blockIdx.x / nTilesN) * 8 + wave;
    const int half = lane >> 4;                  // 0: lanes 0-15, 1: lanes 16-31
    const int l16  = lane & 15;

    // ---- async-stage W[K x 16] panel into LDS: 64 rows / 256-thread pass,
    //      4 lanes x B128 (16B) per row ----
    {
        const int r  = tid >> 2;                 // row within pass
        const int sg = (tid & 3) * 4;            // 4-float column segment
        for (int base = 0; base < K; base += 64) {
            int row = base + r;
            if (row < K) {
                uint64_t gsrc  = (uint64_t)(W + (size_t)row * N + tN * 16 + sg);
                unsigned ldsDst = (unsigned)(uintptr_t)&shW[row * 16 + sg];
                asm volatile("global_load_async_to_lds_b128 %0, %1, off"
                             :: "v"(ldsDst), "v"(gsrc) : "memory");
            }
        }
        asm volatile("s_wait_asynccnt 0x0" ::: "memory");
    }
    __syncthreads();

    const float* Arow = A + (size_t)(tM * 16 + l16) * K + 2 * half;

    v8f acc = {0.f, 0.f, 0.f, 0.f, 0.f, 0.f, 0.f, 0.f};

#if __has_builtin(__builtin_amdgcn_wmma_f32_16x16x4_f32)
    for (int k = 0; k < K; k += 4) {
        v2f af, bf;
        af[0] = Arow[k];
        af[1] = Arow[k + 1];
        bf[0] = shW[(k + 2 * half) * 16 + l16];
        bf[1] = shW[(k + 2 * half + 1) * 16 + l16];
        acc = __builtin_amdgcn_wmma_f32_16x16x4_f32(
                  /*neg_a=*/false, af, /*neg_b=*/false, bf,
                  /*c_mod=*/(short)0, acc, /*reuse_a=*/false, /*reuse_b=*/false);
    }
#else
    for (int k = 0; k < K; k += 4) {             // scalar fallback (unused on gfx1250)
        acc[0] += Arow[k] * shW[(k + 2 * half) * 16 + l16];
    }
#endif

    const int col = tN * 16 + l16;
    float* Yr = Y + (size_t)(tM * 16 + 8 * half) * N + col;
#pragma unroll
    for (int v = 0; v < 8; ++v) {
        float val = acc[v];
        if (ACC) val += Yr[(size_t)v * N];
        Yr[(size_t)v * N] = val;
    }
}

// =====================================================================
// Multi-scale conv1d + BN + ReLU for one kernel size, writes 32-channel
// slice of feat laid out (B*L, 96) row-major.
// =====================================================================
__global__ void msconv_kernel(const float* __restrict__ X,   // (B,3,L)
                              const float* __restrict__ w,   // (32,3,K)
                              const float* __restrict__ bias,
                              const float* __restrict__ bng,
                              const float* __restrict__ bnb,
                              const float* __restrict__ bnm,
                              const float* __restrict__ bnv,
                              float* __restrict__ FEAT,
                              int K, int chOff)
{
    int idx = blockIdx.x * blockDim.x + threadIdx.x;     // BL*32 threads
    if (idx >= BL * 32) return;
    int co  = idx & 31;
    int row = idx >> 5;
    int b   = row >> 12;        // /LEN
    int l   = row & (LEN - 1);
    int pad = K >> 1;
    float acc = bias[co];
    for (int ci = 0; ci < 3; ++ci) {
        const float* xr = X + ((size_t)b * 3 + ci) * LEN;
        const float* wr = w + (size_t)(co * 3 + ci) * K;
        for (int j = 0; j < K; ++j) {
            int pos = l - pad + j;
            if (pos >= 0 && pos < LEN) acc += xr[pos] * wr[j];
        }
    }
    acc = (acc - bnm[co]) * rsqrtf(bnv[co] + LNEPS) * bng[co] + bnb[co];
    FEAT[(size_t)row * CFEAT + chOff + co] = fmaxf(acc, 0.f);
}

// =====================================================================
// LayerNorm over C=96 per row; one wave per row (3 elems/lane, shfl reduce)
// =====================================================================
__global__ void layernorm96_kernel(const float* __restrict__ X,
                                   const float* __restrict__ g,
                                   const float* __restrict__ bta,
                                   float* __restrict__ Y, int rows)
{
    int wave = (blockIdx.x * blockDim.x + threadIdx.x) >> 5;
    int lane = threadIdx.x & 31;
    if (wave >= rows) return;
    const float* xr = X + (size_t)wave * CFEAT;
    float a0 = xr[lane], a1 = xr[lane + 32], a2 = xr[lane + 64];
    float s = a0 + a1 + a2;
    for (int o = 16; o >= 1; o >>= 1) s += __shfl_xor(s, o, 32);
    float mu = s * (1.f / 96.f);
    float d0 = a0 - mu, d1 = a1 - mu, d2 = a2 - mu;
    float q = d0 * d0 + d1 * d1 + d2 * d2;
    for (int o = 16; o >= 1; o >>= 1) q += __shfl_xor(q, o, 32);
    float rstd = rsqrtf(q * (1.f / 96.f) + LNEPS);
    float* yr = Y + (size_t)wave * CFEAT;
    yr[lane]      = d0 * rstd * g[lane]      + bta[lane];
    yr[lane + 32] = d1 * rstd * g[lane + 32] + bta[lane + 32];
    yr[lane + 64] = d2 * rstd * g[lane + 64] + bta[lane + 64];
}

// =====================================================================
// Depthwise causal conv (D_CONV=4) + bias + SiLU.
// xi = first 192 cols of xz (row stride 384); output xc (B*L,192).
// =====================================================================
__global__ void dwconv_silu_kernel(const float* __restrict__ XZ,
                                   const float* __restrict__ w,   // (192,1,4)
                                   const float* __restrict__ b,
                                   float* __restrict__ XC)
{
    int idx = blockIdx.x * blockDim.x + threadIdx.x;   // BL*192
    if (idx >= BL * DI) return;
    int d   = idx % DI;
    int row = idx / DI;
    int l   = row & (LEN - 1);
    float acc = b[d];
#pragma unroll
    for (int j = 0; j < 4; ++j) {
        int pos = l - 3 + j;
        if (pos >= 0) acc += w[d * 4 + j] * XZ[(size_t)(row - 3 + j) * 384 + d];
    }
    XC[idx] = acc / (1.f + __expf(-acc));   // silu
}

// =====================================================================
// Selective scan. 16 lanes per (b,d) pair, one state n per lane.
// dt projected on the fly (6 MACs), per-step shfl reduction over n.
// =====================================================================
__global__ void scan_kernel(const float* __restrict__ DBL,  // (BL,48)
                            const float* __restrict__ XC,   // (BL,192)
                            const float* __restrict__ Wdt,  // (6,192)
                            const float* __restrict__ bdt,  // (192)
                            const float* __restrict__ Alog, // (192,16)
                            float* __restrict__ YS)         // (BL,192)
{
    int n    = threadIdx.x & 15;
    int pair = (blockIdx.x * blockDim.x + threadIdx.x) >> 4;  // 0..1535
    int b = pair / DI, d = pair % DI;
    float Ac = -__expf(Alog[d * NST + n]);
    float w0 = Wdt[0 * DI + d], w1 = Wdt[1 * DI + d], w2 = Wdt[2 * DI + d];
    float w3 = Wdt[3 * DI + d], w4 = Wdt[4 * DI + d], w5 = Wdt[5 * DI + d];
    float bb = bdt[d];
    float h = 0.f;
    const float* dblB = DBL + (size_t)b * LEN * DBLP;
    const float* xcB  = XC  + (size_t)b * LEN * DI + d;
    float*       ysB  = YS  + (size_t)b * LEN * DI + d;
    for (int t = 0; t < LEN; ++t) {
        const float* r = dblB + (size_t)t * DBLP;
        if (t + 8 < LEN) __builtin_prefetch(dblB + (size_t)(t + 8) * DBLP, 0, 0);
        float dtr = bb + r[0]*w0 + r[1]*w1 + r[2]*w2 + r[3]*w3 + r[4]*w4 + r[5]*w5;
        float dt  = (dtr > 20.f) ? dtr : log1pf(__expf(dtr));
        float Bv  = r[DTR + n];
        float Cv  = r[DTR + NST + n];
        float x   = xcB[(size_t)t * DI];
        h = __expf(dt * Ac) * h + dt * x * Bv;
        float y = h * Cv;
        y += __shfl_xor(y, 8, 16);
        y += __shfl_xor(y, 4, 16);
        y += __shfl_xor(y, 2, 16);
        y += __shfl_xor(y, 1, 16);
        if (n == 0) ysB[(size_t)t * DI] = y;
    }
}

// =====================================================================
// Gate: ys = (ys + xc*D) * silu(z),  z = xz[:,192:]
// =====================================================================
__global__ void gate_kernel(float* __restrict__ YS,
                            const float* __restrict__ XC,
                            const float* __restrict__ Dp,
                            const float* __restrict__ XZ)
{
    int idx = blockIdx.x * blockDim.x + threadIdx.x;
    if (idx >= BL * DI) return;
    int d   = idx % DI;
    int row = idx / DI;
    float z = XZ[(size_t)row * 384 + DI + d];
    float y = YS[idx] + XC[idx] * Dp[d];
    YS[idx] = y * (z / (1.f + __expf(-z)));
}

// ============ pack W_x (192x38) -> zero-padded (192x48) ==============
__global__ void pack_wx_kernel(const float* __restrict__ Wx, float* __restrict__ Wp)
{
    int idx = blockIdx.x * blockDim.x + threadIdx.x;   // 192*48
    if (idx >= DI * DBLP) return;
    int r = idx / DBLP, c = idx % DBLP;
    Wp[idx] = (c < (DTR + 2 * NST)) ? Wx[r * (DTR + 2 * NST) + c] : 0.f;
}

// ========== three sigmoid heads (1x1 conv over C=96) =================
__global__ void heads_kernel(const float* __restrict__ H,
                             const float* __restrict__ w0, const float* __restrict__ b0,
                             const float* __restrict__ w1, const float* __restrict__ b1,
                             const float* __restrict__ w2, const float* __restrict__ b2,
                             float* __restrict__ OUT)
{
    int row = blockIdx.x * blockDim.x + threadIdx.x;
    if (row >= BL) return;
    const float* hr = H + (size_t)row * CFEAT;
    float s0 = b0[0], s1 = b1[0], s2 = b2[0];
    for (int c = 0; c < CFEAT; ++c) {
        float h = hr[c];
        s0 += h * w0[c]; s1 += h * w1[c]; s2 += h * w2[c];
    }
    OUT[row]            = 1.f / (1.f + __expf(-s0));
    OUT[BL + row]       = 1.f / (1.f + __expf(-s1));
    OUT[2 * BL + row]   = 1.f / (1.f + __expf(-s2));
}

// ========== rhythm = max over L, one wave per (b,c) ==================
__global__ void rhythm_kernel(const float* __restrict__ H, float* __restrict__ OUT)
{
    int lane = threadIdx.x;                 // block = 32
    int bc   = blockIdx.x;                  // 0..767
    int b = bc / CFEAT, c = bc % CFEAT;
    float mx = -3.402823466e38f;
    for (int l = lane; l < LEN; l += 32)
        mx = fmaxf(mx, H[((size_t)b * LEN + l) * CFEAT + c]);
    for (int o = 16; o >= 1; o >>= 1) mx = fmaxf(mx, __shfl_xor(mx, o, 32));
    if (lane == 0) OUT[(size_t)3 * BL + bc] = mx;
}

// =====================================================================
extern "C" void kernel_launch(void* const* d_in, const int* in_sizes, int n_in,
                              void* d_out, int out_size, void* d_ws, size_t ws_size,
                              hipStream_t stream)
{
    (void)in_sizes; (void)n_in; (void)out_size; (void)ws_size;
    auto IN = [&](int i) { return (const float*)d_in[i]; };

    // input index map (setup_inputs dict order, depth-first)
    // 0: x; 1..18: ms[k] {w,b,bn_g,bn_b,bn_m,bn_v}; 19..29: ssm1; 30..40: ssm2
    // 41: ln_g, 42: ln_b, 43/44 head_qrs, 45/46 head_p, 47/48 head_t
    const float* x = IN(0);
    float* out = (float*)d_out;

    // ---- workspace carve (floats) ----
    float* ws   = (float*)d_ws;
    float* feat = ws;                       // BL*96
    float* hbuf = feat + (size_t)BL * CFEAT;// BL*96
    float* xz   = hbuf + (size_t)BL * CFEAT;// BL*384
    float* xc   = xz   + (size_t)BL * 384;  // BL*192
    float* dbl  = xc   + (size_t)BL * DI;   // BL*48
    float* ys   = dbl  + (size_t)BL * DBLP; // BL*192
    float* wxp  = ys   + (size_t)BL * DI;   // 192*48

    const int TB = 256;

    // ---- multi-scale conv front end ----
    {
        const int ksz[3] = {7, 15, 31};
        int nthr = BL * 32, nblk = (nthr + TB - 1) / TB;
        for (int s = 0; s < 3; ++s) {
            int base = 1 + 6 * s;
            msconv_kernel<<<nblk, TB, 0, stream>>>(
                x, IN(base), IN(base + 1), IN(base + 2), IN(base + 3),
                IN(base + 4), IN(base + 5), feat, ksz[s], 32 * s);
        }
    }

    // ---- two VSS blocks ----
    auto vss = [&](int base) {
        const float* ln_g  = IN(base + 0);
        const float* ln_b  = IN(base + 1);
        const float* W_in  = IN(base + 2);
        const float* convw = IN(base + 3);
        const float* convb = IN(base + 4);
        const float* W_x   = IN(base + 5);
        const float* W_dt  = IN(base + 6);
        const float* b_dt  = IN(base + 7);
        const float* A_log = IN(base + 8);
        const float* Dp    = IN(base + 9);
        const float* W_out = IN(base + 10);

        // LN -> hbuf
        layernorm96_kernel<<<BL / 8, TB, 0, stream>>>(feat, ln_g, ln_b, hbuf, BL);
        // pack W_x with zero pad to 48 cols
        pack_wx_kernel<<<(DI * DBLP + TB - 1) / TB, TB, 0, stream>>>(W_x, wxp);
        // GEMM1: xz = hbuf @ W_in   (32768x96 @ 96x384); 8 waves/block share N-tile
        {
            int nTN = 384 / 16;
            wmma_gemm_f32_kernel<0><<<(BL / 128) * nTN, TB, 0, stream>>>(
                hbuf, W_in, xz, BL, 384, CFEAT, nTN);
        }
        // depthwise causal conv + silu -> xc
        dwconv_silu_kernel<<<(BL * DI + TB - 1) / TB, TB, 0, stream>>>(xz, convw, convb, xc);
        // GEMM2: dbl = xc @ wxpad   (32768x192 @ 192x48)
        {
            int nTN = DBLP / 16;
            wmma_gemm_f32_kernel<0><<<(BL / 128) * nTN, TB, 0, stream>>>(
                xc, wxp, dbl, BL, DBLP, DI, nTN);
        }
        // selective scan -> ys
        scan_kernel<<<(BATCH * DI * NST) / TB, TB, 0, stream>>>(
            dbl, xc, W_dt, b_dt, A_log, ys);
        // gate: ys = (ys + xc*D) * silu(z)
        gate_kernel<<<(BL * DI + TB - 1) / TB, TB, 0, stream>>>(ys, xc, Dp, xz);
        // GEMM3 (+residual): feat += ys @ W_out   (32768x192 @ 192x96)
        {
            int nTN = CFEAT / 16;
            wmma_gemm_f32_kernel<1><<<(BL / 128) * nTN, TB, 0, stream>>>(
                ys, W_out, feat, BL, CFEAT, DI, nTN);
        }
    };
    vss(19);
    vss(30);

    // ---- final LN -> hbuf ----
    layernorm96_kernel<<<BL / 8, TB, 0, stream>>>(feat, IN(41), IN(42), hbuf, BL);

    // ---- heads + rhythm ----
    heads_kernel<<<BL / TB, TB, 0, stream>>>(hbuf, IN(43), IN(44), IN(45), IN(46),
                                             IN(47), IN(48), out);
    rhythm_kernel<<<BATCH * CFEAT, 32, 0, stream>>>(hbuf, out);
}